// WSN_GNN_6992206758516
// MI455X (gfx1250) — compile-verified
//
#include <hip/hip_runtime.h>
#include <hip/hip_bf16.h>

// Problem constants (match reference setup_inputs)
#define GN 50000
#define GE 800000
#define F_IN 128
#define F_E 16
#define GH 64
#define NEG_SLOPE 0.2f

typedef float v2f __attribute__((ext_vector_type(2)));
typedef float v8f __attribute__((ext_vector_type(8)));

// ---------------------------------------------------------------------------
// ea_mean[j] = mean over edges of edge_attr[:, j]   (16 blocks, one per col)
// ---------------------------------------------------------------------------
__global__ void k_ea_mean(const float* __restrict__ ea, float* __restrict__ ea_mean) {
    __shared__ float red[256];
    int j = blockIdx.x, t = threadIdx.x;
    float s = 0.f;
    for (int i = t; i < GE; i += 256) s += ea[(size_t)i * F_E + j];
    red[t] = s; __syncthreads();
    for (int w = 128; w > 0; w >>= 1) { if (t < w) red[t] += red[t + w]; __syncthreads(); }
    if (t == 0) ea_mean[j] = red[0] / (float)GE;
}

// ---------------------------------------------------------------------------
// we[l*16+k] = sum_j We_l[k,j]*ae_l[j];  selfAE[l] = ea_mean . we_l
// ---------------------------------------------------------------------------
__global__ void k_fold_we(const float* __restrict__ We1, const float* __restrict__ ae1,
                          const float* __restrict__ We2, const float* __restrict__ ae2,
                          const float* __restrict__ We3, const float* __restrict__ ae3,
                          const float* __restrict__ ea_mean,
                          float* __restrict__ we, float* __restrict__ selfAE) {
    int t = threadIdx.x;
    if (t < 48) {
        int l = t / 16, k = t % 16;
        const float* We = (l == 0) ? We1 : (l == 1) ? We2 : We3;
        const float* ae = (l == 0) ? ae1 : (l == 1) ? ae2 : ae3;
        float s = 0.f;
        for (int j = 0; j < GH; ++j) s += We[k * GH + j] * ae[j];
        we[t] = s;
    }
    __syncthreads();
    if (t < 3) {
        float s = 0.f;
        for (int k = 0; k < F_E; ++k) s += ea_mean[k] * we[t * 16 + k];
        selfAE[t] = s;
    }
}

// ---------------------------------------------------------------------------
// alphaE[l*E + e] = edge_attr[e] . we_l     (thread per edge)
// ---------------------------------------------------------------------------
__global__ void k_alphaE(const float* __restrict__ ea, const float* __restrict__ we,
                         float* __restrict__ alphaE) {
    int e = blockIdx.x * 256 + threadIdx.x;
    if (e >= GE) return;
    float v[F_E];
    #pragma unroll
    for (int k = 0; k < F_E; ++k) v[k] = ea[(size_t)e * F_E + k];
    #pragma unroll
    for (int l = 0; l < 3; ++l) {
        float s = 0.f;
        #pragma unroll
        for (int k = 0; k < F_E; ++k) s += v[k] * we[l * 16 + k];
        alphaE[(size_t)l * GE + e] = s;
    }
}

// ---------------------------------------------------------------------------
// Zero per-layer accumulators
// ---------------------------------------------------------------------------
__global__ void k_zero(float* __restrict__ out, unsigned* __restrict__ amaxU,
                       float* __restrict__ denom) {
    int idx = blockIdx.x * 256 + threadIdx.x;
    if (idx < GN * GH) out[idx] = 0.f;
    if (idx < GN) { amaxU[idx] = 0u; denom[idx] = 0.f; }
}

// ---------------------------------------------------------------------------
// H[N,64] = A[N,K] @ B[K,64] with V_WMMA_F32_16X16X4_F32.
// Block = 128 thr = 4 waves; wave w -> cols [16w,16w+16); block -> rows [16*bid,..).
// A lane layout (16x4 f32): VGPR0: lanes0-15 K=k+0 / lanes16-31 K=k+2; VGPR1: K=k+1 / k+3.
// B lane layout (4x16 f32): same K split, N = lane%16.
// C/D (16x16 f32): VGPR i: lanes0-15 M=i, lanes16-31 M=8+i; N = lane%16.
// ---------------------------------------------------------------------------
__global__ void k_gemm_wmma(const float* __restrict__ A, const float* __restrict__ B,
                            float* __restrict__ H, int K) {
    const int wave  = threadIdx.x >> 5;
    const int lane  = threadIdx.x & 31;
    const int lh    = lane & 15;          // M (for A) / N (for B,C)
    const int kh    = (lane >> 4) * 2;    // K sub-offset: 0 or 2
    const int m0    = blockIdx.x * 16;
    const int n0    = wave * 16;

    const float* arow = A + (size_t)(m0 + lh) * K;
    v8f c = {};
    for (int k = 0; k < K; k += 4) {
        v2f a, b;
        a.x = arow[k + kh + 0];
        a.y = arow[k + kh + 1];
        b.x = B[(size_t)(k + kh + 0) * GH + n0 + lh];
        b.y = B[(size_t)(k + kh + 1) * GH + n0 + lh];
        c = __builtin_amdgcn_wmma_f32_16x16x4_f32(false, a, false, b,
                                                  (short)0, c, false, false);
    }
    const int rof = (lane >> 4) * 8;
    #pragma unroll
    for (int i = 0; i < 8; ++i)
        H[(size_t)(m0 + rof + i) * GH + n0 + lh] = c[i];
}

// ---------------------------------------------------------------------------
// alpha_src[n] = h[n].a_s ; alpha_dst[n] = h[n].a_d
// ---------------------------------------------------------------------------
__global__ void k_alpha_sd(const float* __restrict__ h, const float* __restrict__ aS,
                           const float* __restrict__ aD, float* __restrict__ alphaS,
                           float* __restrict__ alphaD) {
    int n = blockIdx.x * 256 + threadIdx.x;
    if (n >= GN) return;
    const float* hr = h + (size_t)n * GH;
    float s = 0.f, d = 0.f;
    #pragma unroll 8
    for (int j = 0; j < GH; ++j) { float v = hr[j]; s += v * aS[j]; d += v * aD[j]; }
    alphaS[n] = s; alphaD[n] = d;
}

// Monotone float -> uint mapping for atomicMax-based segment max
__device__ __forceinline__ unsigned f2ord(float f) {
    unsigned u = __float_as_uint(f);
    return (f >= 0.f) ? (u | 0x80000000u) : ~u;
}
__device__ __forceinline__ float ord2f(unsigned u) {
    return (u & 0x80000000u) ? __uint_as_float(u & 0x7FFFFFFFu) : __uint_as_float(~u);
}

// ---------------------------------------------------------------------------
// E1: alpha[e] = leaky(aS[src]+aD[dst]+alphaE[e]);  atomicMax amax[dst]
// ---------------------------------------------------------------------------
__global__ void k_edge_alpha(const long long* __restrict__ src, const long long* __restrict__ dst,
                             const float* __restrict__ alphaE, const float* __restrict__ selfAE,
                             const float* __restrict__ aS, const float* __restrict__ aD,
                             float* __restrict__ alpha, unsigned* __restrict__ amaxU) {
    int e = blockIdx.x * 256 + threadIdx.x;
    const int Etot = GE + GN;
    if (e >= Etot) return;
    int s, d; float ae;
    if (e < GE) { s = (int)src[e]; d = (int)dst[e]; ae = alphaE[e]; }
    else        { s = d = e - GE; ae = *selfAE; }
    float a = aS[s] + aD[d] + ae;
    a = (a > 0.f) ? a : NEG_SLOPE * a;
    alpha[e] = a;
    atomicMax(&amaxU[d], f2ord(a));
}

// ---------------------------------------------------------------------------
// E2: ex = exp(alpha - amax[dst]); alpha[e] <- ex; denom[dst] += ex
// ---------------------------------------------------------------------------
__global__ void k_edge_exp(const long long* __restrict__ dst,
                           const unsigned* __restrict__ amaxU,
                           float* __restrict__ alpha, float* __restrict__ denom) {
    int e = blockIdx.x * 256 + threadIdx.x;
    const int Etot = GE + GN;
    if (e >= Etot) return;
    int d = (e < GE) ? (int)dst[e] : (e - GE);
    float ex = __expf(alpha[e] - ord2f(amaxU[d]));
    alpha[e] = ex;
    atomicAdd(&denom[d], ex);
}

// ---------------------------------------------------------------------------
// E3: out[dst] += h[src] * (ex / (denom[dst]+1e-16))   (one wave32 per edge)
// ---------------------------------------------------------------------------
__global__ void k_edge_scatter(const long long* __restrict__ src, const long long* __restrict__ dst,
                               const float* __restrict__ alpha, const float* __restrict__ denom,
                               const float* __restrict__ h, float* __restrict__ out) {
    int e = (blockIdx.x * blockDim.x + threadIdx.x) >> 5;
    int lane = threadIdx.x & 31;
    const int Etot = GE + GN;
    if (e >= Etot) return;
    int s, d;
    if (e < GE) { s = (int)src[e]; d = (int)dst[e]; }
    else        { s = d = e - GE; }
    float coef = alpha[e] / (denom[d] + 1e-16f);
    const float2 hv = *(const float2*)&h[(size_t)s * GH + lane * 2];
    atomicAdd(&out[(size_t)d * GH + lane * 2 + 0], hv.x * coef);
    atomicAdd(&out[(size_t)d * GH + lane * 2 + 1], hv.y * coef);
}

// ---------------------------------------------------------------------------
// x_next = elu(out + b); nodeEmb (=) or (+=) x_next
// ---------------------------------------------------------------------------
__global__ void k_bias_elu(const float* __restrict__ out, const float* __restrict__ b,
                           float* __restrict__ xnext, float* __restrict__ nodeEmb,
                           int accumulate) {
    int idx = blockIdx.x * 256 + threadIdx.x;
    if (idx >= GN * GH) return;
    float v = out[idx] + b[idx & (GH - 1)];
    v = (v > 0.f) ? v : (__expf(v) - 1.f);
    xnext[idx] = v;
    nodeEmb[idx] = accumulate ? (nodeEmb[idx] + v) : v;
}

// ---------------------------------------------------------------------------
// graph_embedding[j] = mean over nodes of nodeEmb[:, j]  (64 blocks)
// ---------------------------------------------------------------------------
__global__ void k_colmean(const float* __restrict__ nodeEmb, float* __restrict__ gout) {
    __shared__ float red[256];
    int j = blockIdx.x, t = threadIdx.x;
    float s = 0.f;
    for (int n = t; n < GN; n += 256) s += nodeEmb[(size_t)n * GH + j];
    red[t] = s; __syncthreads();
    for (int w = 128; w > 0; w >>= 1) { if (t < w) red[t] += red[t + w]; __syncthreads(); }
    if (t == 0) gout[j] = red[0] / (float)GN;
}

// ---------------------------------------------------------------------------
extern "C" void kernel_launch(void* const* d_in, const int* in_sizes, int n_in,
                              void* d_out, int out_size, void* d_ws, size_t ws_size,
                              hipStream_t stream) {
    const float*     x    = (const float*)d_in[0];
    const long long* ei   = (const long long*)d_in[1];      // int64 per reference
    const float*     ea   = (const float*)d_in[2];
    const long long* srcI = ei;
    const long long* dstI = ei + GE;

    // per-layer params: W, as, ad, We, ae, b  starting at index 3
    const float* W[3];  const float* aS[3]; const float* aD[3];
    const float* We[3]; const float* aeV[3]; const float* bV[3];
    for (int l = 0; l < 3; ++l) {
        W[l]   = (const float*)d_in[3 + 6 * l + 0];
        aS[l]  = (const float*)d_in[3 + 6 * l + 1];
        aD[l]  = (const float*)d_in[3 + 6 * l + 2];
        We[l]  = (const float*)d_in[3 + 6 * l + 3];
        aeV[l] = (const float*)d_in[3 + 6 * l + 4];
        bV[l]  = (const float*)d_in[3 + 6 * l + 5];
    }

    float* nodeEmb = (float*)d_out;                 // [GN, 64]
    float* gEmb    = nodeEmb + (size_t)GN * GH;     // [64]

    // ---- workspace carve-out (256B aligned) ----
    char* ws = (char*)d_ws;
    size_t off = 0;
    auto carve = [&](size_t bytes) -> char* {
        char* p = ws + off;
        off += (bytes + 255) & ~(size_t)255;
        return p;
    };
    float*    h      = (float*)carve((size_t)GN * GH * 4);
    float*    xA     = (float*)carve((size_t)GN * GH * 4);
    float*    xB     = (float*)carve((size_t)GN * GH * 4);
    float*    outAcc = (float*)carve((size_t)GN * GH * 4);
    float*    alpha  = (float*)carve((size_t)(GE + GN) * 4);
    float*    alphaE = (float*)carve((size_t)3 * GE * 4);
    float*    alphS  = (float*)carve((size_t)GN * 4);
    float*    alphD  = (float*)carve((size_t)GN * 4);
    unsigned* amaxU  = (unsigned*)carve((size_t)GN * 4);
    float*    denom  = (float*)carve((size_t)GN * 4);
    float*    eaMean = (float*)carve(16 * 4);
    float*    weF    = (float*)carve(48 * 4);
    float*    selfAE = (float*)carve(3 * 4);
    (void)ws_size; (void)n_in; (void)in_sizes; (void)out_size;

    const int Etot   = GE + GN;
    const int gEdge  = (Etot + 255) / 256;
    const int gNode  = (GN + 255) / 256;
    const int gNH    = (GN * GH + 255) / 256;
    const int gScat  = (Etot + 7) / 8;       // one wave32 per edge, 8 waves/block

    // ---- shared precompute ----
    k_ea_mean<<<F_E, 256, 0, stream>>>(ea, eaMean);
    k_fold_we<<<1, 64, 0, stream>>>(We[0], aeV[0], We[1], aeV[1], We[2], aeV[2],
                                    eaMean, weF, selfAE);
    k_alphaE<<<(GE + 255) / 256, 256, 0, stream>>>(ea, weF, alphaE);

    // ---- three GAT layers ----
    const float* xin = x;
    int K = F_IN;
    float* xbufs[3] = { xA, xB, xA };
    for (int l = 0; l < 3; ++l) {
        float* xnext = xbufs[l];
        k_zero<<<gNH, 256, 0, stream>>>(outAcc, amaxU, denom);
        k_gemm_wmma<<<GN / 16, 128, 0, stream>>>(xin, W[l], h, K);
        k_alpha_sd<<<gNode, 256, 0, stream>>>(h, aS[l], aD[l], alphS, alphD);
        k_edge_alpha<<<gEdge, 256, 0, stream>>>(srcI, dstI, alphaE + (size_t)l * GE,
                                                selfAE + l, alphS, alphD, alpha, amaxU);
        k_edge_exp<<<gEdge, 256, 0, stream>>>(dstI, amaxU, alpha, denom);
        k_edge_scatter<<<gScat, 256, 0, stream>>>(srcI, dstI, alpha, denom, h, outAcc);
        k_bias_elu<<<gNH, 256, 0, stream>>>(outAcc, bV[l], xnext, nodeEmb, l > 0 ? 1 : 0);
        xin = xnext;
        K = GH;
    }

    // ---- graph embedding ----
    k_colmean<<<GH, 256, 0, stream>>>(nodeEmb, gEmb);
}